// WassersteinLoss_45432164057744
// MI455X (gfx1250) — compile-verified
//
#include <hip/hip_runtime.h>
#include <math.h>

// ---------------------------------------------------------------------------
// Sinkhorn divergence (geomloss-style) for two 8192x3 point clouds, gfx1250.
//
// softmin(eps, C, h)_i = -eps * log sum_j exp(h_j - C_ij/eps),
//   C_ij = 0.5*(|x_i|^2 + |y_j|^2) - x_i.y_j
//
// log2-domain with host-folded constants:
//   t2_ij = kdot * d_ij + hy_j
//   d_ij  = x_i.y_j - 0.5*|x_i|^2            (produced entirely by WMMA:
//                                             A K3 slot = -0.5|x_i|^2, B K3 = 1)
//   hy_j  = LOG2E*(logw + pscale*pot_j/eps) - 0.5*LOG2E*|y_j|^2/eps
//   kdot  = LOG2E/eps
//   softmin_i = -eps*LN2 * (max_j t2 + log2 sum_j 2^(t2 - max))
//
// LSE is computed two-pass per 2048-column chunk: pass 1 takes the per-lane
// row max (no exp), pass 2 accumulates 2^(t-max) with ONE v_exp_f32 per
// entry. Chunk partials merge into running (m,s) (2 exps per row per chunk).
//
// V_WMMA_F32_16X16X4_F32 layouts per ISA 7.12.2:
//   A (16x4 f32): lanes 0-15 {v0=K0,v1=K1}, lanes 16-31 {v0=K2,v1=K3}
//   B (4x16 f32): same striping, N on lanes
//   D (16x16 f32, 8 VGPRs): vgpr k: lanes 0-15 -> M=k, lanes 16-31 -> M=8+k,
//                           N = lane&15
// ---------------------------------------------------------------------------

#define NPTS   8192
#define CHUNK  2048
#define LOG2E  1.4426950408889634f
#define LN2    0.6931471805599453f

typedef float v2f __attribute__((ext_vector_type(2)));
typedef float v8f __attribute__((ext_vector_type(8)));

__device__ __forceinline__ float fexp2(float x) { return __builtin_amdgcn_exp2f(x); }
__device__ __forceinline__ float flog2(float x) { return __builtin_amdgcn_logf(x); }

__launch_bounds__(256)
__global__ void softmin_kernel(const float* __restrict__ X,     // row points  [NPTS][3]
                               const float* __restrict__ Y,     // col points  [NPTS][3]
                               const float* __restrict__ pot,   // col potential [NPTS]
                               const float* __restrict__ oldp,  // old row potential [NPTS]
                               float* __restrict__ out,         // new row potential [NPTS]
                               float c0, float c1p, float c2, float kdot,
                               float outbeta, float alpha)
{
    __shared__ float4 tile[CHUNK];   // (y0, y1, y2, hy) per column, 32 KB

    const int lane = threadIdx.x & 31;
    const int wave = threadIdx.x >> 5;
    const int half = lane >> 4;          // 0 -> lanes 0-15, 1 -> lanes 16-31
    const int l15  = lane & 15;
    const int r0   = blockIdx.x * 128 + wave * 16;   // 16-row strip per wave

    // ---- A fragment (row coords; K3 slot carries -0.5*|x_row|^2) ----
    const int ra = r0 + l15;
    const float xa0 = X[ra * 3 + 0], xa1 = X[ra * 3 + 1], xa2 = X[ra * 3 + 2];
    const float xn2 = -0.5f * (xa0 * xa0 + xa1 * xa1 + xa2 * xa2);
    v2f afrag;
    afrag.x = half ? xa2 : xa0;      // K0 / K2
    afrag.y = half ? xn2 : xa1;      // K1 / K3

    // ---- running log2-sum-exp state: 8 rows per lane ----
    float m[8], s[8];
#pragma unroll
    for (int k = 0; k < 8; ++k) { m[k] = -3.0e38f; s[k] = 0.0f; }

    for (int base = 0; base < NPTS; base += CHUNK) {
        // stage column chunk into LDS (coords + folded column bias)
        for (int j = threadIdx.x; j < CHUNK; j += 256) {
            int col = base + j;
            float y0 = Y[col * 3 + 0], y1 = Y[col * 3 + 1], y2c = Y[col * 3 + 2];
            float q  = pot[col];
            float n2 = y0 * y0 + y1 * y1 + y2c * y2c;
            float hy = __builtin_fmaf(c1p, q, __builtin_fmaf(c2, n2, c0));
            tile[j] = make_float4(y0, y1, y2c, hy);
        }
        __syncthreads();

        // ---- pass 1: per-lane row max over this chunk (no exp) ----
        float mc[8];
#pragma unroll
        for (int k = 0; k < 8; ++k) mc[k] = -3.0e38f;

        for (int j0 = 0; j0 < CHUNK; j0 += 16) {
            float4 yc = tile[j0 + l15];
            v2f bfrag;
            bfrag.x = half ? yc.z : yc.x;     // K0 / K2
            bfrag.y = half ? 1.0f : yc.y;     // K1 / K3(=1 -> adds -0.5|x|^2)
            float hy = yc.w;

            v8f acc = {0.f, 0.f, 0.f, 0.f, 0.f, 0.f, 0.f, 0.f};
            v8f d = __builtin_amdgcn_wmma_f32_16x16x4_f32(
                false, afrag, false, bfrag, (short)0, acc, false, false);

#pragma unroll
            for (int k = 0; k < 8; ++k) {
                float t = __builtin_fmaf(kdot, d[k], hy);
                mc[k] = fmaxf(mc[k], t);
            }
        }

        // ---- pass 2: accumulate 2^(t - chunk_max), ONE exp per entry ----
        float sc[8];
#pragma unroll
        for (int k = 0; k < 8; ++k) sc[k] = 0.0f;

        for (int j0 = 0; j0 < CHUNK; j0 += 16) {
            float4 yc = tile[j0 + l15];
            v2f bfrag;
            bfrag.x = half ? yc.z : yc.x;
            bfrag.y = half ? 1.0f : yc.y;
            float hy = yc.w;

            v8f acc = {0.f, 0.f, 0.f, 0.f, 0.f, 0.f, 0.f, 0.f};
            v8f d = __builtin_amdgcn_wmma_f32_16x16x4_f32(
                false, afrag, false, bfrag, (short)0, acc, false, false);

#pragma unroll
            for (int k = 0; k < 8; ++k) {
                float t = __builtin_fmaf(kdot, d[k], hy);
                sc[k] += fexp2(t - mc[k]);
            }
        }

        // ---- merge chunk partials into running (m, s) ----
#pragma unroll
        for (int k = 0; k < 8; ++k) {
            float mn = fmaxf(m[k], mc[k]);
            s[k] = __builtin_fmaf(s[k], fexp2(m[k] - mn), sc[k] * fexp2(mc[k] - mn));
            m[k] = mn;
        }
        __syncthreads();
    }

    // ---- merge partials across the 16 lanes of each half (wave32) ----
#pragma unroll
    for (int k = 0; k < 8; ++k) {
#pragma unroll
        for (int mask = 1; mask < 16; mask <<= 1) {
            float mo = __shfl_xor(m[k], mask, 32);
            float so = __shfl_xor(s[k], mask, 32);
            float mn = fmaxf(m[k], mo);
            s[k] = __builtin_fmaf(s[k], fexp2(m[k] - mn), so * fexp2(mo - mn));
            m[k] = mn;
        }
    }

    // ---- write: out[row] = alpha*oldp[row] + outbeta*(m + log2 s) ----
#pragma unroll
    for (int k = 0; k < 8; ++k) {
        if (l15 == k) {
            int row = r0 + half * 8 + k;
            float v = outbeta * (m[k] + flog2(s[k]));
            out[row] = __builtin_fmaf(alpha, oldp[row], v);
        }
    }
}

__global__ void reduce_kernel(const float* __restrict__ ff, const float* __restrict__ pf,
                              const float* __restrict__ gf, const float* __restrict__ qf,
                              float* __restrict__ out)
{
    __shared__ float sm[256];
    float acc = 0.0f;
    for (int i = threadIdx.x; i < NPTS; i += 256)
        acc += (ff[i] - pf[i]) + (gf[i] - qf[i]);
    sm[threadIdx.x] = acc;
    __syncthreads();
    for (int st = 128; st > 0; st >>= 1) {
        if (threadIdx.x < st) sm[threadIdx.x] += sm[threadIdx.x + st];
        __syncthreads();
    }
    if (threadIdx.x == 0) out[0] = sm[0] * (1.0f / (float)NPTS);
}

extern "C" void kernel_launch(void* const* d_in, const int* in_sizes, int n_in,
                              void* d_out, int out_size, void* d_ws, size_t ws_size,
                              hipStream_t stream)
{
    const float* x = (const float*)d_in[0];
    const float* y = (const float*)d_in[1];
    float* out = (float*)d_out;

    float* ws = (float*)d_ws;
    float* f0 = ws + 0 * NPTS;
    float* g0 = ws + 1 * NPTS;
    float* p0 = ws + 2 * NPTS;
    float* q0 = ws + 3 * NPTS;
    float* f1 = ws + 4 * NPTS;
    float* g1 = ws + 5 * NPTS;
    float* p1 = ws + 6 * NPTS;
    float* q1 = ws + 7 * NPTS;

    const float logw = -logf(8192.0f);   // log(1/N) == log(1/M)

    auto SM = [&](const float* X, const float* Y, const float* pot, float pscale,
                  const float* oldp, float alpha, float beta, float eps, float* o) {
        float inv = 1.0f / eps;
        float c0  = LOG2E * logw;
        float c1p = LOG2E * inv * pscale;
        float c2  = -0.5f * LOG2E * inv;
        float kd  = LOG2E * inv;
        float ob  = -beta * eps * LN2;
        softmin_kernel<<<NPTS / 128, 256, 0, stream>>>(X, Y, pot, oldp, o,
                                                       c0, c1p, c2, kd, ob, alpha);
    };

    // ---- eps schedule (double math, matching the reference) ----
    double sched[128];
    int ns = 0;
    const double ratio  = 0.9 * 0.9;
    const double target = 0.01 * 0.01;
    for (double e = 4.0; e > target && ns < 128; e *= ratio)
        sched[ns++] = e;
    const float eps0      = (float)sched[0];
    const float eps_final = (float)target;

    // ---- init at eps0 with zero duals (pscale=0; pot ptr unused but valid) ----
    SM(x, y, (const float*)y, 0.0f, f0, 0.0f, 1.0f, eps0, f0);
    SM(y, x, (const float*)x, 0.0f, g0, 0.0f, 1.0f, eps0, g0);
    SM(x, x, (const float*)x, 0.0f, p0, 0.0f, 1.0f, eps0, p0);
    SM(y, y, (const float*)y, 0.0f, q0, 0.0f, 1.0f, eps0, q0);

    // ---- symmetric averaged Sinkhorn loop ----
    for (int i = 0; i < ns; ++i) {
        float e = (float)sched[i];
        SM(x, y, g0, 1.0f, f0, 0.5f, 0.5f, e, f1);
        SM(y, x, f0, 1.0f, g0, 0.5f, 0.5f, e, g1);
        SM(x, x, p0, 1.0f, p0, 0.5f, 0.5f, e, p1);
        SM(y, y, q0, 1.0f, q0, 0.5f, 0.5f, e, q1);
        float* t;
        t = f0; f0 = f1; f1 = t;
        t = g0; g0 = g1; g1 = t;
        t = p0; p0 = p1; p1 = t;
        t = q0; q0 = q1; q1 = t;
    }

    // ---- final extrapolation at eps = blur^p (reuse *1 buffers) ----
    SM(x, y, g0, 1.0f, f1, 0.0f, 1.0f, eps_final, f1);
    SM(y, x, f0, 1.0f, g1, 0.0f, 1.0f, eps_final, g1);
    SM(x, x, p0, 1.0f, p1, 0.0f, 1.0f, eps_final, p1);
    SM(y, y, q0, 1.0f, q1, 0.0f, 1.0f, eps_final, q1);

    // ---- debiased divergence: dot(a, f-p) + dot(b, g-q) ----
    reduce_kernel<<<1, 256, 0, stream>>>(f1, p1, g1, q1, out);
}